// GCN_47193100648765
// MI455X (gfx1250) — compile-verified
//
#include <hip/hip_runtime.h>
#include <hip/hip_bf16.h>
#include <math.h>

#define NN 100000
#define EE 1000000
#define GG 64
#define GMM_EPS 1e-15f

typedef __attribute__((ext_vector_type(2))) float v2f;
typedef __attribute__((ext_vector_type(8))) float v8f;

// ---------------------------------------------------------------------------
// Utility kernels
// ---------------------------------------------------------------------------
__global__ void zero_f(float* __restrict__ p, size_t n) {
    size_t i = (size_t)blockIdx.x * blockDim.x + threadIdx.x;
    if (i < n) p[i] = 0.0f;
}

// Pack W = [g | root] : W is K_ x Ncols, g is K_ x gCols, root is K_ x (Ncols-gCols)
__global__ void pack_w(const float* __restrict__ g, const float* __restrict__ root,
                       float* __restrict__ W, int K_, int gCols, int Ncols) {
    int i = blockIdx.x * blockDim.x + threadIdx.x;
    if (i >= K_ * Ncols) return;
    int r = i / Ncols, c = i - r * Ncols;
    W[i] = (c < gCols) ? g[r * gCols + c] : root[r * (Ncols - gCols) + (c - gCols)];
}

// ---------------------------------------------------------------------------
// FP32 WMMA GEMM: Y[nRows x Ncols] = A[nRows x K_] * W[K_ x Ncols]
// One wave per 16x16 output tile, accumulated with V_WMMA_F32_16X16X4_F32.
// nRows must be a multiple of 16 (100000 = 6250*16). K_ multiple of 4.
// ---------------------------------------------------------------------------
__global__ void gemm_wmma_f32(const float* __restrict__ A, const float* __restrict__ W,
                              float* __restrict__ Y, int K_, int Ncols, int ldY,
                              int nTiles, int colTiles) {
    int wid  = threadIdx.x >> 5;
    int tile = blockIdx.x * 4 + wid;          // 4 waves / block
    if (tile >= nTiles) return;               // wave-uniform exit (EXEC all-ones for WMMA)
    int lane = threadIdx.x & 31;
    int rt = tile / colTiles, ct = tile - rt * colTiles;
    int m = lane & 15, half = lane >> 4;

    const float* Arow = A + (size_t)(rt * 16 + m) * (size_t)K_;
    v8f acc = {};
    for (int kb = 0; kb < K_; kb += 4) {
        // A 16x4 f32 fragment: lanes 0-15 hold K=kb..kb+1, lanes 16-31 hold K=kb+2..kb+3
        v2f a;
        a.x = Arow[kb + half * 2 + 0];
        a.y = Arow[kb + half * 2 + 1];
        // B 4x16 f32 fragment: row striped across lanes, K split across lane halves
        const float* Wr = W + (size_t)(kb + half * 2) * (size_t)Ncols + ct * 16 + m;
        v2f b;
        b.x = Wr[0];
        b.y = Wr[Ncols];
        acc = __builtin_amdgcn_wmma_f32_16x16x4_f32(false, a, false, b,
                                                    (short)0, acc, false, false);
    }
    // C/D layout: VGPR v, lanes 0-15 -> row v, lanes 16-31 -> row v+8, col = lane&15
    int rowBase = rt * 16 + half * 8;
    float* Yp = Y + (size_t)rowBase * (size_t)ldY + ct * 16 + m;
#pragma unroll
    for (int v = 0; v < 8; ++v) Yp[(size_t)v * ldY] = acc[v];
}

// ---------------------------------------------------------------------------
// Edge-degree (mean denominator) for dst nodes
// ---------------------------------------------------------------------------
__global__ void deg_kernel(const int* __restrict__ ei, float* __restrict__ deg, int nE) {
    int i = blockIdx.x * blockDim.x + threadIdx.x;
    if (i < nE) unsafeAtomicAdd(&deg[ei[nE + i]], 1.0f);
}

// ---------------------------------------------------------------------------
// GMM edge message + scatter-add, layer 1 (H=32): one wave per edge,
// lane = output channel. Gathers hit the 192MB L2 (proj matrix = 76MB).
// ---------------------------------------------------------------------------
__global__ void edge_msg_h32(const float* __restrict__ Y, const int* __restrict__ ei,
                             const float* __restrict__ ea, const float* __restrict__ mu,
                             const float* __restrict__ sg, float* __restrict__ agg, int nE) {
    int e = blockIdx.x * (blockDim.x >> 5) + (threadIdx.x >> 5);
    if (e >= nE) return;
    int lane = threadIdx.x & 31;
    int s = ei[e], d = ei[nE + e];
    float a0 = ea[e * 3 + 0], a1 = ea[e * 3 + 1], a2 = ea[e * 3 + 2];
    const float* hp = Y + (size_t)s * 192 + lane;
    float msg = 0.0f;
#pragma unroll
    for (int k = 0; k < 5; ++k) {
        float t0 = a0 - mu[k * 3 + 0], t1 = a1 - mu[k * 3 + 1], t2 = a2 - mu[k * 3 + 2];
        float s0 = sg[k * 3 + 0], s1 = sg[k * 3 + 1], s2 = sg[k * 3 + 2];
        float ex = -0.5f * (t0 * t0 / (GMM_EPS + s0 * s0) +
                            t1 * t1 / (GMM_EPS + s1 * s1) +
                            t2 * t2 / (GMM_EPS + s2 * s2));
        msg += __expf(ex) * hp[k * 32];
    }
    unsafeAtomicAdd(&agg[(size_t)d * 32 + lane], msg);
}

// Layer 2 (H=64): lane handles channels {lane, lane+32}
__global__ void edge_msg_h64(const float* __restrict__ Y, const int* __restrict__ ei,
                             const float* __restrict__ ea, const float* __restrict__ mu,
                             const float* __restrict__ sg, float* __restrict__ agg, int nE) {
    int e = blockIdx.x * (blockDim.x >> 5) + (threadIdx.x >> 5);
    if (e >= nE) return;
    int lane = threadIdx.x & 31;
    int s = ei[e], d = ei[nE + e];
    float a0 = ea[e * 3 + 0], a1 = ea[e * 3 + 1], a2 = ea[e * 3 + 2];
    const float* hp = Y + (size_t)s * 384;
    float m0 = 0.0f, m1 = 0.0f;
#pragma unroll
    for (int k = 0; k < 5; ++k) {
        float t0 = a0 - mu[k * 3 + 0], t1 = a1 - mu[k * 3 + 1], t2 = a2 - mu[k * 3 + 2];
        float s0 = sg[k * 3 + 0], s1 = sg[k * 3 + 1], s2 = sg[k * 3 + 2];
        float ex = -0.5f * (t0 * t0 / (GMM_EPS + s0 * s0) +
                            t1 * t1 / (GMM_EPS + s1 * s1) +
                            t2 * t2 / (GMM_EPS + s2 * s2));
        float g = __expf(ex);
        m0 += g * hp[k * 64 + lane];
        m1 += g * hp[k * 64 + lane + 32];
    }
    unsafeAtomicAdd(&agg[(size_t)d * 64 + lane], m0);
    unsafeAtomicAdd(&agg[(size_t)d * 64 + lane + 32], m1);
}

// ---------------------------------------------------------------------------
// out = ELU(agg/max(deg,1) + root + bias), in-place on agg
// ---------------------------------------------------------------------------
__global__ void finalize_elu(float* __restrict__ agg, const float* __restrict__ Yroot,
                             int ldY, const float* __restrict__ deg,
                             const float* __restrict__ bias, int H, int n) {
    int i = blockIdx.x * blockDim.x + threadIdx.x;
    if (i >= n * H) return;
    int ni = i / H, m = i - ni * H;
    float v = agg[i] / fmaxf(deg[ni], 1.0f) + Yroot[(size_t)ni * ldY + m] + bias[m];
    agg[i] = v > 0.0f ? v : expm1f(v);
}

// ---------------------------------------------------------------------------
// Global mean pool: sums + counts
// ---------------------------------------------------------------------------
__global__ void pool_sum(const float* __restrict__ h2, const int* __restrict__ batch,
                         float* __restrict__ pooled, int n) {
    int i = blockIdx.x * blockDim.x + threadIdx.x;
    if (i >= n * 64) return;
    int ni = i >> 6, m = i & 63;
    unsafeAtomicAdd(&pooled[batch[ni] * 64 + m], h2[i]);
}

__global__ void pool_cnt(const int* __restrict__ batch, float* __restrict__ cnt, int n) {
    int i = blockIdx.x * blockDim.x + threadIdx.x;
    if (i < n) unsafeAtomicAdd(&cnt[batch[i]], 1.0f);
}

// ---------------------------------------------------------------------------
// FC head: pooled/cnt -> fc1(ELU) -> fc2 -> log_softmax.  One block, LDS resident.
// ---------------------------------------------------------------------------
__global__ void head_kernel(const float* __restrict__ pooled, const float* __restrict__ cnt,
                            const float* __restrict__ fc1w, const float* __restrict__ fc1b,
                            const float* __restrict__ fc2w, const float* __restrict__ fc2b,
                            float* __restrict__ out) {
    __shared__ float pm[64 * 64];
    __shared__ float z1[64 * 128];
    int tid = threadIdx.x; // 128 threads
    for (int i = tid; i < 64 * 64; i += 128) {
        int g = i >> 6;
        pm[i] = pooled[i] / fmaxf(cnt[g], 1.0f);
    }
    __syncthreads();
    for (int i = tid; i < 64 * 128; i += 128) {
        int g = i >> 7, j = i & 127;
        float acc = fc1b[j];
        for (int k = 0; k < 64; ++k) acc += pm[g * 64 + k] * fc1w[k * 128 + j];
        z1[i] = acc > 0.0f ? acc : expm1f(acc);
    }
    __syncthreads();
    if (tid < 64) {
        float t0 = fc2b[0], t1 = fc2b[1];
        for (int j = 0; j < 128; ++j) {
            float z = z1[tid * 128 + j];
            t0 += z * fc2w[j * 2 + 0];
            t1 += z * fc2w[j * 2 + 1];
        }
        float mx = fmaxf(t0, t1);
        float l = mx + logf(expf(t0 - mx) + expf(t1 - mx));
        out[tid * 2 + 0] = t0 - l;
        out[tid * 2 + 1] = t1 - l;
    }
}

// ---------------------------------------------------------------------------
extern "C" void kernel_launch(void* const* d_in, const int* in_sizes, int n_in,
                              void* d_out, int out_size, void* d_ws, size_t ws_size,
                              hipStream_t stream) {
    const float* x     = (const float*)d_in[0];
    const int*   ei    = (const int*)  d_in[1];
    const float* ea    = (const float*)d_in[2];
    const int*   batch = (const int*)  d_in[3];
    const float* g1    = (const float*)d_in[4];
    const float* mu1   = (const float*)d_in[5];
    const float* sg1   = (const float*)d_in[6];
    const float* root1 = (const float*)d_in[7];
    const float* b1    = (const float*)d_in[8];
    const float* g2    = (const float*)d_in[9];
    const float* mu2   = (const float*)d_in[10];
    const float* sg2   = (const float*)d_in[11];
    const float* root2 = (const float*)d_in[12];
    const float* b2    = (const float*)d_in[13];
    const float* fc1w  = (const float*)d_in[14];
    const float* fc1b  = (const float*)d_in[15];
    const float* fc2w  = (const float*)d_in[16];
    const float* fc2b  = (const float*)d_in[17];
    float* out = (float*)d_out;
    (void)in_sizes; (void)n_in; (void)out_size; (void)ws_size;

    char* ws = (char*)d_ws;
    size_t off = 0;
    auto alloc = [&](size_t bytes) -> float* {
        off = (off + 255) & ~(size_t)255;
        float* p = (float*)(ws + off);
        off += bytes;
        return p;
    };
    float* W1     = alloc((size_t)16 * 192 * 4);          // [g1|root1]
    float* W2     = alloc((size_t)32 * 384 * 4);          // [g2|root2]
    float* Y      = alloc((size_t)NN * 384 * 4);          // projections (layer1 ld=192, layer2 ld=384)
    float* h1     = alloc((size_t)NN * 32 * 4);           // agg1 -> h1 in place
    float* h2     = alloc((size_t)NN * 64 * 4);           // agg2 -> h2 in place
    float* deg    = alloc((size_t)NN * 4);
    float* pooled = alloc((size_t)GG * 64 * 4);
    float* cnt    = alloc((size_t)GG * 4);

    // --- init: packed weights + zero accumulators ---
    pack_w<<<(16 * 192 + 255) / 256, 256, 0, stream>>>(g1, root1, W1, 16, 160, 192);
    pack_w<<<(32 * 384 + 255) / 256, 256, 0, stream>>>(g2, root2, W2, 32, 320, 384);
    zero_f<<<((size_t)NN * 32 + 255) / 256, 256, 0, stream>>>(h1, (size_t)NN * 32);
    zero_f<<<((size_t)NN * 64 + 255) / 256, 256, 0, stream>>>(h2, (size_t)NN * 64);
    zero_f<<<(NN + 255) / 256, 256, 0, stream>>>(deg, (size_t)NN);
    zero_f<<<(GG * 64 + 255) / 256, 256, 0, stream>>>(pooled, (size_t)GG * 64);
    zero_f<<<1, 64, 0, stream>>>(cnt, (size_t)GG);

    deg_kernel<<<(EE + 255) / 256, 256, 0, stream>>>(ei, deg, EE);

    // --- layer 1 ---
    {
        int colTiles = 192 / 16, nTiles = (NN / 16) * colTiles;      // 6250*12
        gemm_wmma_f32<<<(nTiles + 3) / 4, 128, 0, stream>>>(x, W1, Y, 16, 192, 192, nTiles, colTiles);
    }
    edge_msg_h32<<<(EE + 7) / 8, 256, 0, stream>>>(Y, ei, ea, mu1, sg1, h1, EE);
    finalize_elu<<<((size_t)NN * 32 + 255) / 256, 256, 0, stream>>>(h1, Y + 160, 192, deg, b1, 32, NN);

    // --- layer 2 ---
    {
        int colTiles = 384 / 16, nTiles = (NN / 16) * colTiles;      // 6250*24
        gemm_wmma_f32<<<(nTiles + 3) / 4, 128, 0, stream>>>(h1, W2, Y, 32, 384, 384, nTiles, colTiles);
    }
    edge_msg_h64<<<(EE + 7) / 8, 256, 0, stream>>>(Y, ei, ea, mu2, sg2, h2, EE);
    finalize_elu<<<((size_t)NN * 64 + 255) / 256, 256, 0, stream>>>(h2, Y + 320, 384, deg, b2, 64, NN);

    // --- pool + head ---
    pool_cnt<<<(NN + 255) / 256, 256, 0, stream>>>(batch, cnt, NN);
    pool_sum<<<((size_t)NN * 64 + 255) / 256, 256, 0, stream>>>(h2, batch, pooled, NN);
    head_kernel<<<1, 128, 0, stream>>>(pooled, cnt, fc1w, fc1b, fc2w, fc2b, out);
}